// SemanticBranch_31164282699948
// MI455X (gfx1250) — compile-verified
//
#include <hip/hip_runtime.h>
#include <hip/hip_bf16.h>

// ---------------------------------------------------------------------------
// DGCNN semantic branch for MI455X (gfx1250, wave32, WMMA).
// All GEMMs use V_WMMA_F32_16X16X4_F32 (full fp32 precision, CDNA5 matrix path).
// Pipeline:
//   S1  : y1 = w1 @ x + b1            (WMMA GEMM, 256x1216x8192 x2)
//   BN1 : channel stats -> scale/shift -> h1 (c-major) + h1T (n-major)
//   KNN : top-20 neighbor indices (tiled LDS, per-thread sorted top-20)
//   YC  : yc = (we_b - we_a) @ h1 + be (WMMA GEMM)
//   EDGE: yn = we_a @ h1T[idx]  (gather WMMA GEMM); per-(n,o) max/min over k
//         and channel sum/sumsq reduced in LDS (edgeconv never materialized).
//   SEL : h2 = leakyrelu(scale * (scale>=0 ? max : min) + shift)
//   S2,S3, final: WMMA GEMMs + BN stages, out = wf @ h4 + bf.
// ---------------------------------------------------------------------------

typedef float v2f __attribute__((ext_vector_type(2)));
typedef float v8f __attribute__((ext_vector_type(8)));

#define NEG_SLOPE 0.2f
#define BN_EPS    1e-5f

__device__ __forceinline__ float leaky(float v) { return v >= 0.f ? v : NEG_SLOPE * v; }

// Monotone float<->uint encoding so atomicMax/atomicMin(u32) order like floats.
__device__ __forceinline__ unsigned fenc(float f) {
    unsigned b = __float_as_uint(f);
    return (b & 0x80000000u) ? ~b : (b | 0x80000000u);
}
__device__ __forceinline__ float fdec(unsigned k) {
    unsigned b = (k & 0x80000000u) ? (k & 0x7fffffffu) : ~k;
    return __uint_as_float(b);
}

// ---------------------------------------------------------------------------
// Generic fp32 WMMA GEMM:  Y[b][o][n] = sum_c W[o][c] * X[b][c][n] + bias[o]
// Block: 256 threads (8 waves). Block tile 64(M) x 128(N); wave tile 32x32
// = 2x2 WMMA accumulators (A reused across 2 col tiles, B across 2 row tiles).
// ---------------------------------------------------------------------------
__device__ __forceinline__ void store_tile(float* __restrict__ Yb, v8f acc,
                                           int row0, int col0, int M, int N,
                                           const float* __restrict__ bias,
                                           bool hi, int l16) {
    int col = col0 + l16;
#pragma unroll
    for (int v = 0; v < 8; ++v) {
        int row = row0 + v + (hi ? 8 : 0);
        if (row < M) Yb[(size_t)row * N + col] = acc[v] + bias[row];
    }
}

__global__ void gemm_wmma(const float* __restrict__ W, int ldw,
                          const float* __restrict__ X, size_t xStride,
                          float* __restrict__ Y, size_t yStride,
                          const float* __restrict__ bias,
                          int M, int C, int N) {
    const int b = blockIdx.z;
    const float* Xb = X + (size_t)b * xStride;
    float* Yb = Y + (size_t)b * yStride;

    const int tid  = threadIdx.x;
    const int wave = tid >> 5;
    const int lane = tid & 31;
    const int l16  = lane & 15;
    const bool hi  = lane >= 16;
    const int cOff = hi ? 2 : 0;

    const int m0 = blockIdx.y * 64 + (wave & 1) * 32;
    const int n0 = blockIdx.x * 128 + (wave >> 1) * 32;

    int rowA0 = m0 + l16;       if (rowA0 > M - 1) rowA0 = M - 1;
    int rowA1 = m0 + 16 + l16;  if (rowA1 > M - 1) rowA1 = M - 1;

    const float* a0p = W + (size_t)rowA0 * ldw + cOff;
    const float* a1p = W + (size_t)rowA1 * ldw + cOff;
    const float* b0p = Xb + (size_t)cOff * N + n0 + l16;
    const float* b1p = b0p + 16;

    const v8f z = {0, 0, 0, 0, 0, 0, 0, 0};
    v8f acc00 = z, acc01 = z, acc10 = z, acc11 = z;

    for (int c0 = 0; c0 < C; c0 += 4) {
        v2f a0 = *(const v2f*)(a0p + c0);          // contiguous pair -> b64
        v2f a1 = *(const v2f*)(a1p + c0);
        v2f b0, b1;
        b0.x = b0p[(size_t)c0 * N];  b0.y = b0p[(size_t)(c0 + 1) * N];
        b1.x = b1p[(size_t)c0 * N];  b1.y = b1p[(size_t)(c0 + 1) * N];
        acc00 = __builtin_amdgcn_wmma_f32_16x16x4_f32(false, a0, false, b0, (short)0, acc00, false, false);
        acc01 = __builtin_amdgcn_wmma_f32_16x16x4_f32(false, a0, false, b1, (short)0, acc01, false, false);
        acc10 = __builtin_amdgcn_wmma_f32_16x16x4_f32(false, a1, false, b0, (short)0, acc10, false, false);
        acc11 = __builtin_amdgcn_wmma_f32_16x16x4_f32(false, a1, false, b1, (short)0, acc11, false, false);
    }
    store_tile(Yb, acc00, m0,      n0,      M, N, bias, hi, l16);
    store_tile(Yb, acc01, m0,      n0 + 16, M, N, bias, hi, l16);
    store_tile(Yb, acc10, m0 + 16, n0,      M, N, bias, hi, l16);
    store_tile(Yb, acc11, m0 + 16, n0 + 16, M, N, bias, hi, l16);
}

// --------------------------- BN stats / apply ------------------------------
__global__ void chan_stats(const float* __restrict__ Y, size_t bstride,
                           int M, int N, int B, float* __restrict__ sums) {
    const int o = blockIdx.x, tid = threadIdx.x;
    float s = 0.f, q = 0.f;
    for (int b = 0; b < B; ++b) {
        const float* row = Y + (size_t)b * bstride + (size_t)o * N;
        for (int n = tid; n < N; n += 256) { float v = row[n]; s += v; q += v * v; }
    }
    __shared__ float rs[256], rq[256];
    rs[tid] = s; rq[tid] = q; __syncthreads();
    for (int st = 128; st > 0; st >>= 1) {
        if (tid < st) { rs[tid] += rs[tid + st]; rq[tid] += rq[tid + st]; }
        __syncthreads();
    }
    if (tid == 0) { sums[o] = rs[0]; sums[M + o] = rq[0]; }
}

__global__ void finalize_ss(const float* __restrict__ sums,
                            const float* __restrict__ g, const float* __restrict__ bt,
                            float count, int M, float* __restrict__ ss) {
    const int o = threadIdx.x;
    if (o < M) {
        float mean = sums[o] / count;
        float var  = sums[M + o] / count - mean * mean;
        float sc   = g[o] * rsqrtf(var + BN_EPS);
        ss[o]     = sc;
        ss[M + o] = bt[o] - mean * sc;
    }
}

__global__ void norm_act(const float* __restrict__ Y, size_t bstride,
                         const float* __restrict__ ss, int M, int N,
                         float* __restrict__ outC, float* __restrict__ outT) {
    const int o = blockIdx.x, b = blockIdx.z;
    const int n = blockIdx.y * 256 + threadIdx.x;
    float v = Y[(size_t)b * bstride + (size_t)o * N + n];
    float r = leaky(ss[o] * v + ss[M + o]);
    outC[((size_t)b * M + o) * N + n] = r;
    if (outT) outT[((size_t)b * N + n) * M + o] = r;   // n-major copy for gathers
}

__global__ void zero512(float* s) {
    int i = blockIdx.x * 256 + threadIdx.x;
    if (i < 512) s[i] = 0.f;
}

// ------------------------------- kNN top-20 --------------------------------
__global__ void knn_topk(const float* __restrict__ p, int* __restrict__ idxG, int N) {
    const int b = blockIdx.y;
    const int n = blockIdx.x * 256 + threadIdx.x;
    const float* pb = p + (size_t)b * 3 * N;
    const float px = pb[n], py = pb[N + n], pz = pb[2 * N + n];
    const float sqn = px * px + py * py + pz * pz;

    __shared__ float tx[256], ty[256], tz[256], tq[256];
    float bd[20]; int bi[20];
#pragma unroll
    for (int t = 0; t < 20; ++t) { bd[t] = -3.4e38f; bi[t] = 0; }

    for (int tb = 0; tb < N; tb += 256) {
        __syncthreads();
        const int j = threadIdx.x;
        float cx = pb[tb + j], cy = pb[N + tb + j], cz = pb[2 * N + tb + j];
        tx[j] = cx; ty[j] = cy; tz[j] = cz; tq[j] = cx * cx + cy * cy + cz * cz;
        __syncthreads();
        for (int j2 = 0; j2 < 256; ++j2) {
            float nd = 2.f * (px * tx[j2] + py * ty[j2] + pz * tz[j2]) - sqn - tq[j2];
            if (nd > bd[19]) {                 // insert into descending sorted list
                bd[19] = nd; bi[19] = tb + j2;
#pragma unroll
                for (int t = 19; t >= 1; --t) {
                    if (bd[t] > bd[t - 1]) {
                        float td = bd[t]; bd[t] = bd[t - 1]; bd[t - 1] = td;
                        int   ti = bi[t]; bi[t] = bi[t - 1]; bi[t - 1] = ti;
                    }
                }
            }
        }
    }
    int* out = idxG + ((size_t)b * N + n) * 20;
#pragma unroll
    for (int t = 0; t < 20; ++t) out[t] = bi[t];
}

// ------------------------- edgeconv helpers --------------------------------
__global__ void prep_wd(const float* __restrict__ we, float* __restrict__ wd) {
    const int o = blockIdx.x, c = threadIdx.x;
    wd[o * 256 + c] = we[(size_t)o * 512 + 256 + c] - we[(size_t)o * 512 + c];
}

// Gather WMMA GEMM: yn[o,(n,k)] = we_a[o,:] . h1T[idx[n,k],:]; fuse +yc, then
// reduce max/min over k and channel sum/sumsq in LDS. Block owns 16 n x 64 o.
__global__ void edge_wmma(const float* __restrict__ we, const float* __restrict__ h1T,
                          const int* __restrict__ idxG, const float* __restrict__ yc,
                          float* __restrict__ emx, float* __restrict__ emn,
                          float* __restrict__ sums, int N) {
    const int b = blockIdx.z;
    const int nbase = blockIdx.x * 16;
    const int obase = blockIdx.y * 64;
    const int tid = threadIdx.x;

    __shared__ int sIdx[320];                 // 16 n x 20 k
    __shared__ unsigned sMx[1024], sMn[1024]; // 64 o x 16 n encoded max/min
    __shared__ float sS[64], sQ[64];          // per-o partial sum / sumsq

    for (int i = tid; i < 320; i += 256)
        sIdx[i] = idxG[((size_t)b * N + nbase) * 20 + i];   // contiguous block
    for (int i = tid; i < 1024; i += 256) { sMx[i] = 0u; sMn[i] = 0xFFFFFFFFu; }
    if (tid < 64) { sS[tid] = 0.f; sQ[tid] = 0.f; }
    __syncthreads();

    const int wave = tid >> 5, lane = tid & 31;
    const int l16 = lane & 15;
    const bool hi = lane >= 16;
    const int cOff = hi ? 2 : 0;
    const int ot = wave & 3;
    const int orow = obase + ot * 16 + l16;
    const float* wrow = we + (size_t)orow * 512 + cOff;     // we_a (cols 0..255)

    for (int j = 0; j < 10; ++j) {
        const int ct = (wave >> 2) + 2 * j;     // waves 0-3: even tiles, 4-7: odd
        const int col = ct * 16 + l16;
        const int g = sIdx[col];
        const float* bp = h1T + ((size_t)b * N + g) * 256 + cOff;  // contiguous row
        v8f acc = {0, 0, 0, 0, 0, 0, 0, 0};
        for (int c0 = 0; c0 < 256; c0 += 4) {
            v2f a  = *(const v2f*)(wrow + c0);
            v2f bb = *(const v2f*)(bp + c0);
            acc = __builtin_amdgcn_wmma_f32_16x16x4_f32(false, a, false, bb, (short)0, acc, false, false);
        }
        const int nloc = col / 20;
        const int myN = nbase + nloc;
#pragma unroll
        for (int v = 0; v < 8; ++v) {
            const int o = obase + ot * 16 + v + (hi ? 8 : 0);
            float y = acc[v] + yc[((size_t)b * 256 + o) * N + myN];
            const int slot = (o - obase) * 16 + nloc;
            unsigned key = fenc(y);
            atomicMax(&sMx[slot], key);
            atomicMin(&sMn[slot], key);
            atomicAdd(&sS[o - obase], y);
            atomicAdd(&sQ[o - obase], y * y);
        }
    }
    __syncthreads();

    for (int i = tid; i < 1024; i += 256) {
        const int ol = i >> 4, nl = i & 15;
        const size_t off = ((size_t)b * 256 + obase + ol) * N + nbase + nl;
        emx[off] = fdec(sMx[i]);
        emn[off] = fdec(sMn[i]);
    }
    if (tid < 64) {
        atomicAdd(&sums[obase + tid], sS[tid]);
        atomicAdd(&sums[256 + obase + tid], sQ[tid]);
    }
}

// h2 = leakyrelu(scale * (scale>=0 ? max_k y : min_k y) + shift)
__global__ void edge_select(const float* __restrict__ emx, const float* __restrict__ emn,
                            const float* __restrict__ ss, float* __restrict__ h2, int N) {
    const int o = blockIdx.x, b = blockIdx.z;
    const int n = blockIdx.y * 256 + threadIdx.x;
    const size_t off = ((size_t)b * 256 + o) * N + n;
    float sc = ss[o];
    float y = sc >= 0.f ? emx[off] : emn[off];
    h2[off] = leaky(sc * y + ss[256 + o]);
}

// ---------------------------------------------------------------------------
extern "C" void kernel_launch(void* const* d_in, const int* in_sizes, int n_in,
                              void* d_out, int out_size, void* d_ws, size_t ws_size,
                              hipStream_t stream) {
    (void)in_sizes; (void)n_in; (void)out_size; (void)ws_size;
    const float* x   = (const float*)d_in[0];
    const float* p   = (const float*)d_in[1];
    const float* w1  = (const float*)d_in[2];
    const float* b1  = (const float*)d_in[3];
    const float* g1  = (const float*)d_in[4];
    const float* bt1 = (const float*)d_in[5];
    const float* we  = (const float*)d_in[6];
    const float* be  = (const float*)d_in[7];
    const float* ge  = (const float*)d_in[8];
    const float* bte = (const float*)d_in[9];
    const float* w2  = (const float*)d_in[10];
    const float* b2  = (const float*)d_in[11];
    const float* g2  = (const float*)d_in[12];
    const float* bt2 = (const float*)d_in[13];
    const float* w3  = (const float*)d_in[14];
    const float* b3  = (const float*)d_in[15];
    const float* g3  = (const float*)d_in[16];
    const float* bt3 = (const float*)d_in[17];
    const float* wf  = (const float*)d_in[18];
    const float* bf  = (const float*)d_in[19];

    const int N = 8192;
    char* wsb = (char*)d_ws;
    const size_t SZ = (size_t)2 * 256 * N * sizeof(float);   // 16 MB per buffer
    float* A0 = (float*)(wsb);
    float* A1 = (float*)(wsb + SZ);
    float* A2 = (float*)(wsb + 2 * SZ);
    float* A3 = (float*)(wsb + 3 * SZ);
    int*   idxb = (int*)(wsb + 4 * SZ);                       // 1.31 MB
    float* wd   = (float*)(wsb + 4 * SZ + (size_t)2 * 1024 * 1024);
    float* sums = wd + 256 * 256;
    float* ss   = sums + 512;

    const dim3 blk(256);

    // S1: y1 = w1 @ x + b1  -> A0
    gemm_wmma<<<dim3(64, 4, 2), blk, 0, stream>>>(w1, 1216, x, (size_t)1216 * N,
                                                  A0, (size_t)256 * N, b1, 256, 1216, N);
    chan_stats<<<dim3(256, 1, 1), blk, 0, stream>>>(A0, (size_t)256 * N, 256, N, 2, sums);
    finalize_ss<<<1, blk, 0, stream>>>(sums, g1, bt1, (float)(2 * N), 256, ss);
    norm_act<<<dim3(256, 32, 2), blk, 0, stream>>>(A0, (size_t)256 * N, ss, 256, N,
                                                   A1 /*h1 c-major*/, A2 /*h1T n-major*/);
    // kNN (independent of above)
    knn_topk<<<dim3(32, 2), blk, 0, stream>>>(p, idxb, N);

    // yc = (we_b - we_a) @ h1 + be  -> A0
    prep_wd<<<dim3(256), blk, 0, stream>>>(we, wd);
    gemm_wmma<<<dim3(64, 4, 2), blk, 0, stream>>>(wd, 256, A1, (size_t)256 * N,
                                                  A0, (size_t)256 * N, be, 256, 256, N);
    // edgeconv gather-GEMM + k-reduction: emx->A3, emn->A1
    zero512<<<2, blk, 0, stream>>>(sums);
    edge_wmma<<<dim3(512, 4, 2), blk, 0, stream>>>(we, A2, idxb, A0, A3, A1, sums, N);
    finalize_ss<<<1, blk, 0, stream>>>(sums, ge, bte, (float)(2 * N * 20), 256, ss);
    edge_select<<<dim3(256, 32, 2), blk, 0, stream>>>(A3, A1, ss, A0 /*h2*/, N);

    // S2: y2 = w2 @ h2 + b2 -> A1 ; h3 -> A2
    gemm_wmma<<<dim3(64, 4, 2), blk, 0, stream>>>(w2, 256, A0, (size_t)256 * N,
                                                  A1, (size_t)256 * N, b2, 256, 256, N);
    chan_stats<<<dim3(256, 1, 1), blk, 0, stream>>>(A1, (size_t)256 * N, 256, N, 2, sums);
    finalize_ss<<<1, blk, 0, stream>>>(sums, g2, bt2, (float)(2 * N), 256, ss);
    norm_act<<<dim3(256, 32, 2), blk, 0, stream>>>(A1, (size_t)256 * N, ss, 256, N, A2, nullptr);

    // S3: y3 = w3 @ h3 + b3 -> A3 (M=128) ; h4 -> A0
    gemm_wmma<<<dim3(64, 2, 2), blk, 0, stream>>>(w3, 256, A2, (size_t)256 * N,
                                                  A3, (size_t)128 * N, b3, 128, 256, N);
    chan_stats<<<dim3(128, 1, 1), blk, 0, stream>>>(A3, (size_t)128 * N, 128, N, 2, sums);
    finalize_ss<<<1, blk, 0, stream>>>(sums, g3, bt3, (float)(2 * N), 128, ss);
    norm_act<<<dim3(128, 32, 2), blk, 0, stream>>>(A3, (size_t)128 * N, ss, 128, N, A0, nullptr);

    // final: out = wf @ h4 + bf (M=50, row-guarded)
    gemm_wmma<<<dim3(64, 1, 2), blk, 0, stream>>>(wf, 128, A0, (size_t)128 * N,
                                                  (float*)d_out, (size_t)50 * N, bf, 50, 128, N);
}